// EF_58360015618072
// MI455X (gfx1250) — compile-verified
//
#include <hip/hip_runtime.h>
#include <cstdint>
#include <cstddef>

// Problem constants (match reference)
#define B_SZ  32
#define CIN   2048
#define CMID  1024
#define HH    14
#define WW    14
#define MPAD  512          // 32 batches * 16 padded h-rows
#define NCLS  30

typedef __attribute__((ext_vector_type(2))) float v2f;
typedef __attribute__((ext_vector_type(4))) float v4f;
typedef __attribute__((ext_vector_type(8))) float v8f;

// ---------------------------------------------------------------------------
// Kernel A: K-paired reduced f1.
//   ATP[k2][mp][2] (k2 = k/2, mp = b*16+hp) with ATP[k2][mp][p] =
//     sum_w f1[b, 2*k2+p, hp, w]   (hp = 14,15 -> zero pad rows)
// Thread t writes word (t & 1023) of row (t >> 10): fully contiguous stores.
// ---------------------------------------------------------------------------
__global__ void __launch_bounds__(256)
reduce_f1_kernel(const float* __restrict__ f1, float* __restrict__ ATP) {
    uint32_t t  = blockIdx.x * 256u + threadIdx.x;   // 2048 * 512 threads
    uint32_t p  = t & 1u;                            // channel parity
    uint32_t mp = (t >> 1) & 511u;                   // b*16 + hp
    uint32_t hp = mp & 15u;
    uint32_t b  = mp >> 4;
    uint32_t i  = ((t >> 10) << 1) + p;              // input channel
    float s = 0.0f;
    if (hp < 14u) {
        const float* q = f1 + (((size_t)b * CIN + i) * HH + hp) * WW;
        #pragma unroll
        for (int w = 0; w < WW; ++w) s += q[w];
    }
    ATP[((size_t)(t >> 10) << 10) + (t & 1023u)] = s;
}

// ---------------------------------------------------------------------------
// Kernel B: Y[b*16+hp][c] = sum_w f2[b,c,hp,w]   (hp=14,15 -> 0 padding)
// ---------------------------------------------------------------------------
__global__ void __launch_bounds__(256)
reduce_f2_kernel(const float* __restrict__ f2, float* __restrict__ Y) {
    uint32_t t  = blockIdx.x * 256u + threadIdx.x;   // 1024 * 512 threads
    uint32_t hp = t & 15u;
    uint32_t b  = (t >> 4) & 31u;
    uint32_t c  = t >> 9;
    float s = 0.0f;
    if (hp < 14u) {
        const float* q = f2 + (((size_t)b * CMID + c) * HH + hp) * WW;
        #pragma unroll
        for (int w = 0; w < WW; ++w) s += q[w];
    }
    Y[(size_t)(b * 16u + hp) * CMID + c] = s;
}

// ---------------------------------------------------------------------------
// Kernel C: fused WMMA GEMM + h-pool epilogue.
//   grid.x = 64 channel tiles (16 each), grid.y = 4 batch groups of 8 waves.
//   Each wave: one batch b, one 16-ch tile, K=2048 via v_wmma_f32_16x16x4.
//   A fragments: one global_load_b64 per WMMA (K-paired layout).
//   B tile staged in LDS: one b128 load + one ds_store_b128 per thread.
//   P[b][c] = (1/2744) * sum_h X[b*16+h][c] * Y[b*16+h][c]
// ---------------------------------------------------------------------------
__global__ void __launch_bounds__(256)
gemm_pool_kernel(const float* __restrict__ ATP,
                 const float* __restrict__ conv_w,
                 const float* __restrict__ Y,
                 float* __restrict__ P) {
    __shared__ float Wt[16 * 68];                    // padded stride: bank-safe

    const uint32_t tid  = threadIdx.x;
    const uint32_t lane = tid & 31u;
    const uint32_t wave = tid >> 5;
    const uint32_t half = lane >> 4;                 // 0: lanes 0-15, 1: 16-31
    const uint32_t lm   = lane & 15u;
    const uint32_t cb   = blockIdx.x * 16u;          // channel-tile base
    const uint32_t b    = blockIdx.y * 8u + wave;    // batch owned by wave

    // Staging coords: thread -> (row r of W-tile, 4-float column group)
    const uint32_t sr = tid >> 4;                    // 0..15
    const uint32_t sc = (tid & 15u) << 2;            // 0,4,...,60
    const float* wsrc = conv_w + (size_t)(cb + sr) * CIN + sc;

    const v2f* apairs = (const v2f*)ATP;             // [k2][mp] of float2

    v8f acc = {};

    for (uint32_t kb = 0; kb < CIN; kb += 64u) {
        // Stage conv_w[cb..cb+15][kb..kb+63] -> LDS: one b128 each way.
        v4f wv = *(const v4f*)(wsrc + kb);
        *(v4f*)&Wt[sr * 68u + sc] = wv;
        // Branchless speculative prefetch of the next chunk (clamped).
        uint32_t kpf = (kb + 64u < CIN) ? (kb + 64u) : kb;
        __builtin_prefetch(wsrc + kpf, 0, 0);
        __syncthreads();

        // A pairs for this wave: k2 = (kb+kk)/2 + half, mp = b*16+lm
        const v2f* abase = apairs + ((size_t)(kb >> 1) + half) * MPAD
                                  + b * 16u + lm;
        const float* wrow = &Wt[lm * 68u + 2u * half];

        #pragma unroll
        for (int kk = 0; kk < 64; kk += 4) {
            v2f a = abase[(size_t)(kk >> 1) * MPAD]; // K = kb+kk+2*half+{0,1}
            v2f bf;
            bf.x = wrow[kk];                         // B[K][n=lm]
            bf.y = wrow[kk + 1];
            acc = __builtin_amdgcn_wmma_f32_16x16x4_f32(
                false, a, false, bf, (short)0, acc, false, false);
        }
        __syncthreads();
    }

    // Epilogue: lane L, VGPR r holds X[M = 8*half + r][N = lm] for tile b.
    // Pool over h (pad rows of Y are zero): dot with Y, cross-half combine.
    float s = 0.0f;
    const float* yrow = Y + (size_t)(b * 16u + half * 8u) * CMID + cb + lm;
    #pragma unroll
    for (int r = 0; r < 8; ++r) s += acc[r] * yrow[(size_t)r * CMID];
    s += __shfl_down(s, 16);
    if (half == 0u)
        P[(size_t)b * CMID + cb + lm] = s * (1.0f / 2744.0f);  // 1/(H*W*W)
}

// ---------------------------------------------------------------------------
// Kernel D: FC head. out[b][n] = P[b]·fc_w[n] + fc_b[n]. One wave per (n,b).
// ---------------------------------------------------------------------------
__global__ void __launch_bounds__(32)
fc_kernel(const float* __restrict__ P, const float* __restrict__ fc_w,
          const float* __restrict__ fc_b, float* __restrict__ out) {
    uint32_t n = blockIdx.x, b = blockIdx.y, lane = threadIdx.x;
    float s = 0.0f;
    for (uint32_t c = lane; c < CMID; c += 32u)
        s += P[(size_t)b * CMID + c] * fc_w[(size_t)n * CMID + c];
    #pragma unroll
    for (int off = 16; off > 0; off >>= 1) s += __shfl_down(s, off);
    if (lane == 0u) out[(size_t)b * NCLS + n] = s + fc_b[n];
}

// ---------------------------------------------------------------------------
extern "C" void kernel_launch(void* const* d_in, const int* in_sizes, int n_in,
                              void* d_out, int out_size, void* d_ws, size_t ws_size,
                              hipStream_t stream) {
    const float* f1     = (const float*)d_in[0];  // [32,2048,14,14]
    const float* f2     = (const float*)d_in[1];  // [32,1024,14,14]
    const float* conv_w = (const float*)d_in[2];  // [1024,2048]
    const float* fc_w   = (const float*)d_in[3];  // [30,1024]
    const float* fc_b   = (const float*)d_in[4];  // [30]
    float* out = (float*)d_out;                   // [32,30]

    // Workspace layout (all fp32):
    //   ATP : [1024][512][2] = 4 MiB   (K-paired reduced f1, h-padded)
    //   Y   : [512][1024]    = 2 MiB   (reduced f2, h-padded)
    //   P   : [32][1024]     = 128 KiB (pooled bilinear features)
    char* ws = (char*)d_ws;
    float* ATP = (float*)(ws);
    float* Y   = (float*)(ws + (size_t)CIN * MPAD * sizeof(float));
    float* P   = (float*)(ws + (size_t)CIN * MPAD * sizeof(float)
                             + (size_t)MPAD * CMID * sizeof(float));

    reduce_f1_kernel<<<(CIN * MPAD) / 256, 256, 0, stream>>>(f1, ATP);
    reduce_f2_kernel<<<(CMID * MPAD) / 256, 256, 0, stream>>>(f2, Y);
    gemm_pool_kernel<<<dim3(CMID / 16, B_SZ / 8), 256, 0, stream>>>(ATP, conv_w, Y, P);
    fc_kernel<<<dim3(NCLS, B_SZ), 32, 0, stream>>>(P, fc_w, fc_b, out);
}